// SOT_84284438216987
// MI455X (gfx1250) — compile-verified
//
#include <hip/hip_runtime.h>
#include <math.h>

// Sinkhorn OT on MI455X (gfx1250, wave32).
// Design: M,K stored fp32 (K = 67MB -> L2-resident across 100 iterations),
// u/v/accumulations fp64 for range. Cost GEMM via V_WMMA_F32_16X16X4_F32.

#define BATCH 64
#define NPT   512
#define DIMD  128
#define REGF  0.01f
#define TOLD  1e-5
#define NFBLK 1024

typedef __attribute__((ext_vector_type(2))) float v2f;
typedef __attribute__((ext_vector_type(8))) float v8f;

// ---------------- init ----------------
__global__ void sot_init(double* u, double* v, double* up, double* vp,
                         unsigned* mmax, int* done, unsigned long long* maxdiff) {
  int t = blockIdx.x * blockDim.x + threadIdx.x;
  if (t < BATCH * NPT) {
    double iv = 1.0 / NPT;
    u[t] = iv; v[t] = iv; up[t] = iv; vp[t] = iv;
  }
  if (t < BATCH) mmax[t] = 0u;
  if (t == 0) { *done = 0; *maxdiff = 0ull; }
}

// ---------------- squared norms ----------------
__global__ void sot_sqnorm(const float* __restrict__ x, float* __restrict__ sq) {
  int t = blockIdx.x * blockDim.x + threadIdx.x;
  if (t >= BATCH * NPT) return;
  const float4* p = (const float4*)(x + (size_t)t * DIMD);
  float s = 0.f;
#pragma unroll
  for (int d = 0; d < DIMD / 4; ++d) {
    float4 q = p[d];
    s += q.x * q.x + q.y * q.y + q.z * q.z + q.w * q.w;
  }
  sq[t] = s;
}

// ---------------- cost matrix via f32 WMMA ----------------
// grid (8, 4, 64): block computes a 64x128 tile; 8 waves, each wave 2x2 WMMA tiles.
__global__ void __launch_bounds__(256)
sot_cost_wmma(const float* __restrict__ x1, const float* __restrict__ x2,
              const float* __restrict__ sq1, const float* __restrict__ sq2,
              float* __restrict__ M, unsigned* __restrict__ Mmax) {
  const int b    = blockIdx.z;
  const int lane = threadIdx.x & 31;
  const int wave = threadIdx.x >> 5;
  const int i0 = blockIdx.x * 64  + (wave & 1) * 32;
  const int j0 = blockIdx.y * 128 + (wave >> 1) * 32;

  const float* A  = x1 + (size_t)b * NPT * DIMD;
  const float* Bm = x2 + (size_t)b * NPT * DIMD;

  const int lm = lane & 15;
  const int kh = (lane >> 4) * 2;      // A/B frag: lane L holds K = kh, kh+1

  v8f c00 = {}; v8f c01 = {}; v8f c10 = {}; v8f c11 = {};
  for (int k = 0; k < DIMD; k += 4) {
    v2f a0 = *(const v2f*)(A  + (size_t)(i0 + lm)      * DIMD + k + kh);
    v2f a1 = *(const v2f*)(A  + (size_t)(i0 + 16 + lm) * DIMD + k + kh);
    v2f b0 = *(const v2f*)(Bm + (size_t)(j0 + lm)      * DIMD + k + kh);
    v2f b1 = *(const v2f*)(Bm + (size_t)(j0 + 16 + lm) * DIMD + k + kh);
    c00 = __builtin_amdgcn_wmma_f32_16x16x4_f32(false, a0, false, b0, (short)0, c00, false, false);
    c01 = __builtin_amdgcn_wmma_f32_16x16x4_f32(false, a0, false, b1, (short)0, c01, false, false);
    c10 = __builtin_amdgcn_wmma_f32_16x16x4_f32(false, a1, false, b0, (short)0, c10, false, false);
    c11 = __builtin_amdgcn_wmma_f32_16x16x4_f32(false, a1, false, b1, (short)0, c11, false, false);
  }

  // C layout: VGPR r -> lanes 0-15: M=r, N=lane; lanes 16-31: M=r+8, N=lane-16
  const int rowoff = (lane >> 4) * 8;
  const int cn = lane & 15;
  float* Mb = M + (size_t)b * NPT * NPT;
  const float* s1 = sq1 + b * NPT;
  const float* s2 = sq2 + b * NPT;
  float lmax = 0.f;
#pragma unroll
  for (int r = 0; r < 8; ++r) {
    int gi0 = i0 + r + rowoff, gi1 = i0 + 16 + r + rowoff;
    int gj0 = j0 + cn,         gj1 = j0 + 16 + cn;
    float m;
    m = sqrtf(fmaxf(s1[gi0] + s2[gj0] - 2.f * c00[r], 0.f));
    Mb[(size_t)gi0 * NPT + gj0] = m; lmax = fmaxf(lmax, m);
    m = sqrtf(fmaxf(s1[gi0] + s2[gj1] - 2.f * c01[r], 0.f));
    Mb[(size_t)gi0 * NPT + gj1] = m; lmax = fmaxf(lmax, m);
    m = sqrtf(fmaxf(s1[gi1] + s2[gj0] - 2.f * c10[r], 0.f));
    Mb[(size_t)gi1 * NPT + gj0] = m; lmax = fmaxf(lmax, m);
    m = sqrtf(fmaxf(s1[gi1] + s2[gj1] - 2.f * c11[r], 0.f));
    Mb[(size_t)gi1 * NPT + gj1] = m; lmax = fmaxf(lmax, m);
  }
  for (int off = 16; off; off >>= 1) lmax = fmaxf(lmax, __shfl_xor(lmax, off, 32));
  if (lane == 0) atomicMax(&Mmax[b], __float_as_uint(lmax));   // M >= 0 -> bits monotone
}

// ---------------- K = exp(-M / (reg * Mmax)) ----------------
__global__ void sot_expk(const float* __restrict__ M, const unsigned* __restrict__ Mmax,
                         float* __restrict__ K) {
  int b = blockIdx.y;
  size_t base = (size_t)b * NPT * NPT;
  int idx = blockIdx.x * blockDim.x + threadIdx.x;      // grid.x = 1024, 256 thr
  float inv = -1.0f / (REGF * __uint_as_float(Mmax[b]));
  K[base + idx] = expf(M[base + idx] * inv);
}

// ---------------- v = b / (K^T u) ----------------
// grid (4, 64), 128 thr; column j per thread, row-coalesced loads, u in LDS.
__global__ void __launch_bounds__(128)
sot_ktu_v(const float* __restrict__ K, const double* __restrict__ u,
          double* __restrict__ v, int* __restrict__ done) {
  if (*done) return;
  __shared__ double us[NPT];
  int b = blockIdx.y;
  int j = blockIdx.x * 128 + threadIdx.x;
  const double* ub = u + b * NPT;
  for (int t = threadIdx.x; t < NPT; t += 128) us[t] = ub[t];
  __syncthreads();
  const float* Kb = K + (size_t)b * NPT * NPT + j;
  double acc = 0.0;
#pragma unroll 8
  for (int i = 0; i < NPT; ++i) acc += (double)Kb[(size_t)i * NPT] * us[i];
  double vv = (1.0 / NPT) / acc;
  if (acc == 0.0 || !isfinite(vv)) atomicOr(done, 1);
  v[b * NPT + j] = vv;
}

// ---------------- u = a / (K v) ----------------
// grid (64, 64), 256 thr; one row per wave, float4 loads, shuffle reduce.
__global__ void __launch_bounds__(256)
sot_kv_u(const float* __restrict__ K, const double* __restrict__ v,
         double* __restrict__ u, int* __restrict__ done) {
  if (*done) return;
  __shared__ double vs[NPT];
  int b = blockIdx.y;
  int lane = threadIdx.x & 31, wave = threadIdx.x >> 5;
  const double* vb = v + b * NPT;
  for (int t = threadIdx.x; t < NPT; t += 256) vs[t] = vb[t];
  __syncthreads();
  int i = blockIdx.x * 8 + wave;
  const float4* Kr = (const float4*)(K + (size_t)b * NPT * NPT + (size_t)i * NPT);
  double acc = 0.0;
#pragma unroll
  for (int t = 0; t < 4; ++t) {
    int j4 = lane + 32 * t;
    float4 kk = Kr[j4];
    int j = j4 * 4;
    acc += (double)kk.x * vs[j] + (double)kk.y * vs[j + 1]
         + (double)kk.z * vs[j + 2] + (double)kk.w * vs[j + 3];
  }
  for (int off = 16; off; off >>= 1) acc += __shfl_xor(acc, off, 32);
  if (lane == 0) {
    double uu = (1.0 / NPT) / acc;
    if (!isfinite(uu)) atomicOr(done, 1);
    u[b * NPT + i] = uu;
  }
}

// ---------------- convergence machinery ----------------
__global__ void sot_reset_maxdiff(unsigned long long* maxdiff) { *maxdiff = 0ull; }

__global__ void __launch_bounds__(256)
sot_pdiff(const float* __restrict__ K, const double* __restrict__ u,
          const double* __restrict__ v, const double* __restrict__ up,
          const double* __restrict__ vp, const int* __restrict__ done,
          unsigned long long* __restrict__ maxdiff) {
  if (*done) return;
  size_t stride = (size_t)gridDim.x * blockDim.x;
  size_t total = (size_t)BATCH * NPT * NPT;
  double lm = 0.0;
  for (size_t idx = (size_t)blockIdx.x * blockDim.x + threadIdx.x; idx < total; idx += stride) {
    size_t b = idx >> 18, r = idx & 262143, i = r >> 9, j = r & 511;
    double d = fabs((double)K[idx] * (u[b * NPT + i] * v[b * NPT + j]
                                    - up[b * NPT + i] * vp[b * NPT + j]));
    lm = fmax(lm, d);
  }
  for (int off = 16; off; off >>= 1) lm = fmax(lm, __shfl_xor(lm, off, 32));
  if ((threadIdx.x & 31) == 0)
    atomicMax(maxdiff, (unsigned long long)__double_as_longlong(lm));
}

__global__ void sot_commit(const double* __restrict__ u, const double* __restrict__ v,
                           double* __restrict__ up, double* __restrict__ vp,
                           const int* __restrict__ done) {
  if (*done) return;
  int t = blockIdx.x * blockDim.x + threadIdx.x;
  if (t < BATCH * NPT) { up[t] = u[t]; vp[t] = v[t]; }
}

__global__ void sot_check(int* done, const unsigned long long* maxdiff) {
  if (!*done && __longlong_as_double((long long)(*maxdiff)) < TOLD) *done = 1;
}

// ---------------- final sum(M * P), P = up * K * vp^T ----------------
__global__ void __launch_bounds__(256)
sot_final_partials(const float* __restrict__ M, const float* __restrict__ K,
                   const double* __restrict__ up, const double* __restrict__ vp,
                   double* __restrict__ partials) {
  __shared__ double sacc[8];
  size_t stride = (size_t)gridDim.x * blockDim.x;
  size_t total = (size_t)BATCH * NPT * NPT;
  double acc = 0.0;
  for (size_t idx = (size_t)blockIdx.x * blockDim.x + threadIdx.x; idx < total; idx += stride) {
    size_t b = idx >> 18, r = idx & 262143, i = r >> 9, j = r & 511;
    acc += (double)M[idx] * (double)K[idx] * up[b * NPT + i] * vp[b * NPT + j];
  }
  for (int off = 16; off; off >>= 1) acc += __shfl_xor(acc, off, 32);
  int lane = threadIdx.x & 31, wave = threadIdx.x >> 5;
  if (lane == 0) sacc[wave] = acc;
  __syncthreads();
  if (threadIdx.x == 0) {
    double s = 0.0;
    for (int w = 0; w < 8; ++w) s += sacc[w];
    partials[blockIdx.x] = s;
  }
}

__global__ void sot_final_sum(const double* __restrict__ partials, float* __restrict__ out) {
  double s = 0.0;
  for (int i = 0; i < NFBLK; ++i) s += partials[i];
  *out = (float)s;
}

// ---------------- host launcher ----------------
extern "C" void kernel_launch(void* const* d_in, const int* in_sizes, int n_in,
                              void* d_out, int out_size, void* d_ws, size_t ws_size,
                              hipStream_t stream) {
  (void)in_sizes; (void)n_in; (void)out_size; (void)ws_size;
  const float* x1 = (const float*)d_in[0];
  const float* x2 = (const float*)d_in[1];
  float* out = (float*)d_out;

  // workspace carve-out (~136 MB total)
  char* ws = (char*)d_ws;
  size_t off = 0;
  auto carve = [&](size_t bytes) -> char* {
    char* p = ws + off;
    off += bytes; off = (off + 255) & ~(size_t)255;
    return p;
  };
  float*  Mbuf = (float*)carve((size_t)BATCH * NPT * NPT * sizeof(float));
  float*  Kbuf = (float*)carve((size_t)BATCH * NPT * NPT * sizeof(float));
  float*  sq1  = (float*)carve((size_t)BATCH * NPT * sizeof(float));
  float*  sq2  = (float*)carve((size_t)BATCH * NPT * sizeof(float));
  unsigned* Mmax = (unsigned*)carve(BATCH * sizeof(unsigned));
  double* u  = (double*)carve((size_t)BATCH * NPT * sizeof(double));
  double* v  = (double*)carve((size_t)BATCH * NPT * sizeof(double));
  double* up = (double*)carve((size_t)BATCH * NPT * sizeof(double));
  double* vp = (double*)carve((size_t)BATCH * NPT * sizeof(double));
  int* done = (int*)carve(sizeof(int));
  unsigned long long* maxdiff = (unsigned long long*)carve(sizeof(unsigned long long));
  double* partials = (double*)carve((size_t)NFBLK * sizeof(double));

  const int vecBlocks = (BATCH * NPT + 255) / 256;
  sot_init<<<vecBlocks, 256, 0, stream>>>(u, v, up, vp, Mmax, done, maxdiff);
  sot_sqnorm<<<vecBlocks, 256, 0, stream>>>(x1, sq1);
  sot_sqnorm<<<vecBlocks, 256, 0, stream>>>(x2, sq2);

  sot_cost_wmma<<<dim3(8, 4, BATCH), 256, 0, stream>>>(x1, x2, sq1, sq2, Mbuf, Mmax);
  sot_expk<<<dim3(1024, BATCH), 256, 0, stream>>>(Mbuf, Mmax, Kbuf);

  for (int it = 0; it < 100; ++it) {
    sot_ktu_v<<<dim3(4, BATCH), 128, 0, stream>>>(Kbuf, u, v, done);
    sot_kv_u<<<dim3(64, BATCH), 256, 0, stream>>>(Kbuf, v, u, done);
    if ((it + 1) % 10 == 0) {
      sot_reset_maxdiff<<<1, 1, 0, stream>>>(maxdiff);
      sot_pdiff<<<2048, 256, 0, stream>>>(Kbuf, u, v, up, vp, done, maxdiff);
      sot_commit<<<vecBlocks, 256, 0, stream>>>(u, v, up, vp, done);
      sot_check<<<1, 1, 0, stream>>>(done, maxdiff);
    }
  }

  sot_final_partials<<<NFBLK, 256, 0, stream>>>(Mbuf, Kbuf, up, vp, partials);
  sot_final_sum<<<1, 1, 0, stream>>>(partials, out);
}